// TeleportAttention_1975684956488
// MI455X (gfx1250) — compile-verified
//
#include <hip/hip_runtime.h>
#include <stdint.h>

// ---------------------------------------------------------------------------
// TeleportAttention KV-cache FP8-range quantize + circular-buffer append.
// Bandwidth-bound (~257 MiB @ 23.3 TB/s ~= 11 us floor). One workgroup per
// token (S blocks x 256 threads, 8 wave32 waves).
// CDNA5 paths: global_load_async_to_lds_b128 (ASYNCcnt), s_wait_asynccnt,
// wave32 shuffle reduction, non-temporal b128 streaming stores.
// ---------------------------------------------------------------------------

typedef float v4f __attribute__((ext_vector_type(4)));

#define FP8_MAX_F     448.0f
#define FP8_HI_F      224.0f
#define UNC_THRESH_F  0.1f
#define EPS_F         1e-8f

__device__ __forceinline__ float amax4(v4f a) {
    return fmaxf(fmaxf(fabsf(a.x), fabsf(a.y)), fmaxf(fabsf(a.z), fabsf(a.w)));
}

__device__ __forceinline__ v4f quant4(v4f a, float inv) {
    v4f q;
    q.x = fminf(fmaxf(a.x * inv, -FP8_MAX_F), FP8_MAX_F);
    q.y = fminf(fmaxf(a.y * inv, -FP8_MAX_F), FP8_MAX_F);
    q.z = fminf(fmaxf(a.z * inv, -FP8_MAX_F), FP8_MAX_F);
    q.w = fminf(fmaxf(a.w * inv, -FP8_MAX_F), FP8_MAX_F);
    return q;
}

// CDNA5 async copy: global memory -> LDS, 16 bytes per lane, tracked by ASYNCcnt.
__device__ __forceinline__ void async_ld_b128(uint32_t lds_off, uint64_t gaddr) {
    asm volatile("global_load_async_to_lds_b128 %0, %1, off"
                 :: "v"(lds_off), "v"(gaddr)
                 : "memory");
}

__device__ __forceinline__ void wait_asynccnt0() {
    asm volatile("s_wait_asynccnt 0x0" ::: "memory");
}

template <int H, int D, int NT>
__global__ __launch_bounds__(NT) void teleport_quant_append_kernel(
    const float* __restrict__ k,        // [S, H, D] (B==1)
    const float* __restrict__ v,        // [S, H, D]
    const float* __restrict__ unc,      // [S]
    const int*   __restrict__ cur_pos,  // scalar
    float* __restrict__ k_out,          // [H, M, D]
    float* __restrict__ v_out,          // [H, M, D]
    float* __restrict__ ks_out,         // [H, M]
    float* __restrict__ vs_out,         // [H, M]
    int M)
{
    constexpr int HD    = H * D;   // floats per token per tensor (4096)
    constexpr int C4    = HD / 4;  // float4 chunks per tensor (1024)
    constexpr int CPT   = C4 / NT; // chunks per thread (4) -- compile-time trips
    constexpr int NWAVE = NT / 32; // wave32
    static_assert(C4 % NT == 0, "tile must divide evenly across the block");

    __shared__ __align__(16) float lds_k[HD];   // 16 KiB
    __shared__ __align__(16) float lds_v[HD];   // 16 KiB
    __shared__ float red_k[NWAVE];
    __shared__ float red_v[NWAVE];

    const int s   = blockIdx.x;
    const int tid = threadIdx.x;

    const float* kg = k + (size_t)s * HD;
    const float* vg = v + (size_t)s * HD;

    // ---- stage token tiles into LDS via async DMA path (no exec guards) ----
    #pragma unroll
    for (int j = 0; j < CPT; ++j) {
        const int c = j * NT + tid;
        async_ld_b128((uint32_t)(uintptr_t)(lds_k + 4 * c),
                      (uint64_t)(uintptr_t)(kg + 4 * c));
        async_ld_b128((uint32_t)(uintptr_t)(lds_v + 4 * c),
                      (uint64_t)(uintptr_t)(vg + 4 * c));
    }
    wait_asynccnt0();     // this wave's async copies landed in LDS
    __syncthreads();      // all waves' copies visible workgroup-wide

    // ---- per-token amax over H*D (pass 1: LDS reads) ----
    float kmax = 0.0f, vmax = 0.0f;
    #pragma unroll
    for (int j = 0; j < CPT; ++j) {
        const int c = j * NT + tid;
        kmax = fmaxf(kmax, amax4(*(const v4f*)(lds_k + 4 * c)));
        vmax = fmaxf(vmax, amax4(*(const v4f*)(lds_v + 4 * c)));
    }
    // wave32 butterfly reduction
    #pragma unroll
    for (int m = 16; m >= 1; m >>= 1) {
        kmax = fmaxf(kmax, __shfl_xor(kmax, m, 32));
        vmax = fmaxf(vmax, __shfl_xor(vmax, m, 32));
    }
    if ((tid & 31) == 0) {
        red_k[tid >> 5] = kmax;
        red_v[tid >> 5] = vmax;
    }
    __syncthreads();
    float ka = red_k[0], va = red_v[0];
    #pragma unroll
    for (int i = 1; i < NWAVE; ++i) {
        ka = fmaxf(ka, red_k[i]);
        va = fmaxf(va, red_v[i]);
    }

    // ---- per-token scale ----
    const float div = (unc[s] > UNC_THRESH_F) ? FP8_HI_F : FP8_MAX_F;
    const float ks  = fmaxf(ka / div, EPS_F);
    const float vs  = fmaxf(va / div, EPS_F);
    const float kinv = 1.0f / ks;
    const float vinv = 1.0f / vs;

    const int pos = (cur_pos[0] + s) % M;

    // ---- quantize + scatter into [H, M, D] caches (pass 2: LDS reads) ----
    #pragma unroll
    for (int j = 0; j < CPT; ++j) {
        const int c = j * NT + tid;
        const int e = 4 * c;
        const int h = e / D;
        const int d = e % D;          // multiple of 4 -> 16B aligned stores
        v4f qa = quant4(*(const v4f*)(lds_k + e), kinv);
        v4f qb = quant4(*(const v4f*)(lds_v + e), vinv);
        float* dk = k_out + ((size_t)h * M + pos) * D + d;
        float* dv = v_out + ((size_t)h * M + pos) * D + d;
        __builtin_nontemporal_store(qa, (v4f*)dk);   // streaming: write-only here
        __builtin_nontemporal_store(qb, (v4f*)dv);
    }

    // ---- broadcast scales to all heads ----
    if (tid < H) {
        __builtin_nontemporal_store(ks, ks_out + (size_t)tid * M + pos);
    } else if (tid < 2 * H) {
        __builtin_nontemporal_store(vs, vs_out + (size_t)(tid - H) * M + pos);
    }
}

extern "C" void kernel_launch(void* const* d_in, const int* in_sizes, int n_in,
                              void* d_out, int out_size, void* d_ws, size_t ws_size,
                              hipStream_t stream) {
    (void)n_in; (void)out_size; (void)d_ws; (void)ws_size;

    constexpr int H = 32, D = 128, NT = 256;

    const float* k        = (const float*)d_in[0];
    const float* v        = (const float*)d_in[1];
    const float* unc      = (const float*)d_in[2];
    const float* k_cache  = (const float*)d_in[3];
    const float* v_cache  = (const float*)d_in[4];
    const float* k_scales = (const float*)d_in[5];
    const float* v_scales = (const float*)d_in[6];
    const int*   cur_pos  = (const int*)d_in[7];

    const int S = in_sizes[2];                       // B*S with B==1
    const long cache_elems = (long)in_sizes[3];      // H*M*D
    const int M = (int)(cache_elems / ((long)H * D));
    const long scale_elems = (long)in_sizes[5];      // H*M

    float* out    = (float*)d_out;
    float* k_out  = out;
    float* v_out  = out + cache_elems;
    float* ks_out = out + 2 * cache_elems;
    float* vs_out = out + 2 * cache_elems + scale_elems;

    // If the append does not cover every circular-buffer slot, seed the output
    // with the previous cache contents (graph-capture-safe async D2D copies).
    // With S >= M every residue of (current_pos + i) % M is overwritten.
    if (S < M) {
        hipMemcpyAsync(k_out,  k_cache,  cache_elems * sizeof(float), hipMemcpyDeviceToDevice, stream);
        hipMemcpyAsync(v_out,  v_cache,  cache_elems * sizeof(float), hipMemcpyDeviceToDevice, stream);
        hipMemcpyAsync(ks_out, k_scales, scale_elems * sizeof(float), hipMemcpyDeviceToDevice, stream);
        hipMemcpyAsync(vs_out, v_scales, scale_elems * sizeof(float), hipMemcpyDeviceToDevice, stream);
    }

    dim3 grid((unsigned)S), block(NT);
    teleport_quant_append_kernel<H, D, NT><<<grid, block, 0, stream>>>(
        k, v, unc, cur_pos, k_out, v_out, ks_out, vs_out, M);
}